// DGL_Net_31181462569288
// MI455X (gfx1250) — compile-verified
//
#include <hip/hip_runtime.h>
#include <hip/hip_bf16.h>
#include <math.h>

// ---------------------------------------------------------------------------
// GCN (DGL GraphConv x3, norm='both') for MI455X / gfx1250.
//  - fp32 WMMA (v_wmma_f32_16x16x4_f32) for the node-feature GEMMs
//  - weights staged in LDS K-pair-interleaved so every B fragment is a single
//    aligned ds_load_b64 into an even VGPR pair (no shuffle moves)
//  - A-fragment loads clamped (not predicated) + software-pipelined so
//    global_load_b64 overlaps the WMMA chain instead of stalling on loadcnt
//  - edge gather/scatter-add with global_atomic_add_f32 (L2-resident buffers)
// ---------------------------------------------------------------------------

typedef float v2f __attribute__((ext_vector_type(2)));
typedef float v8f __attribute__((ext_vector_type(8)));

#define N_NODES 100000
#define N_EDGES 1600000

// ---------------------------------------------------------------- fill zero
__global__ __launch_bounds__(256) void fill_zero_kernel(float* __restrict__ p, int n) {
    int i = blockIdx.x * 256 + threadIdx.x;
    if (i < n) p[i] = 0.0f;
}

// ---------------------------------------------------------------- degrees
__global__ __launch_bounds__(256) void degree_kernel(const int* __restrict__ src,
                                                     const int* __restrict__ dst,
                                                     float* __restrict__ deg_out,
                                                     float* __restrict__ deg_in,
                                                     int nedges) {
    int e = blockIdx.x * 256 + threadIdx.x;
    if (e < nedges) {
        unsafeAtomicAdd(&deg_out[src[e]], 1.0f);
        unsafeAtomicAdd(&deg_in[dst[e]], 1.0f);
    }
}

// deg -> clip(deg,1)^-0.5, in place (applied to both deg arrays at once)
__global__ __launch_bounds__(256) void invsqrt_kernel(float* __restrict__ v, int n) {
    int i = blockIdx.x * 256 + threadIdx.x;
    if (i < n) {
        float d = v[i];
        d = d < 1.0f ? 1.0f : d;
        v[i] = 1.0f / sqrtf(d);
    }
}

// ---------------------------------------------------------------- WMMA GEMM
// H[row, :] = (X[row, :] * scale[row]) @ W      (X: nrows x CIN, W: CIN x COUT)
// Block = 256 threads = 8 waves; block covers 128 rows; wave covers 16 rows x COUT.
//
// LDS layout (K-pair interleaved): Wpair[p*COUT + n] = { W[2p][n], W[2p+1][n] }.
// A lane needing B K-pair (k+2*hi, k+1+2*hi) reads Wpair[(k/2 + hi)*COUT + n]
// as one 8-byte ds_load_b64 -> lands directly in an even-aligned VGPR pair.
template <int CIN, int COUT>
__global__ __launch_bounds__(256) void gemm_scaled_wmma(const float* __restrict__ X,
                                                        const float* __restrict__ scale,
                                                        const float* __restrict__ W,
                                                        float* __restrict__ H,
                                                        int nrows) {
    __shared__ float2 Wpair[(CIN / 2) * COUT];

    const int tid = threadIdx.x;
    // cooperative interleaved stage of W into LDS
    for (int i = tid; i < (CIN / 2) * COUT; i += 256) {
        int p = i / COUT;              // K-pair index
        int n = i - p * COUT;          // column
        float2 w;
        w.x = W[(2 * p + 0) * COUT + n];
        w.y = W[(2 * p + 1) * COUT + n];
        Wpair[i] = w;
    }
    __syncthreads();

    const int wave = tid >> 5;
    const int lane = tid & 31;
    const int lo = lane & 15;       // M (for A/stores) or N (for B) index
    const int hi = lane >> 4;       // selects K pair {0,1} vs {2,3}
    const int m0 = blockIdx.x * 128 + wave * 16;
    const int arow = m0 + lo;       // logical row this lane feeds into A
    const bool rok = arow < nrows;
    // Clamp (never branch): OOB lanes load a valid row but scale s==0 kills it.
    const int crow = rok ? arow : (nrows - 1);
    const float s = rok ? scale[arow] : 0.0f;
    const float* xrow = X + (size_t)crow * CIN + 2 * hi;
    // per-lane B base: K-pair offset 'hi', column lo (tile offset added per t)
    const float2* bbase = Wpair + hi * COUT + lo;

    constexpr int NT = COUT / 16;   // number of 16-wide column tiles
    v8f zero = {};
    v8f acc[NT];
#pragma unroll
    for (int t = 0; t < NT; ++t) acc[t] = zero;

    // Software-pipelined A fragment: prefetch k+4 before the k WMMA chain.
    v2f a_nxt;
    {
        float2 av = *reinterpret_cast<const float2*>(xrow);
        a_nxt.x = av.x * s;
        a_nxt.y = av.y * s;
    }

#pragma unroll
    for (int k = 0; k < CIN; k += 4) {
        v2f a = a_nxt;
        if (k + 4 < CIN) {  // compile-time after full unroll
            float2 av = *reinterpret_cast<const float2*>(xrow + k + 4);
            a_nxt.x = av.x * s;
            a_nxt.y = av.y * s;
        }
        const float2* brow = bbase + (k / 2) * COUT;
#pragma unroll
        for (int t = 0; t < NT; ++t) {
            float2 bv = brow[t * 16];   // single aligned ds_load_b64
            v2f b;
            b.x = bv.x;
            b.y = bv.y;
            acc[t] = __builtin_amdgcn_wmma_f32_16x16x4_f32(
                false, a, false, b, (short)0, acc[t], false, false);
        }
    }

    // D layout: VGPR r -> (M = m0 + r + 8*hi, N = t*16 + lo)
#pragma unroll
    for (int t = 0; t < NT; ++t) {
#pragma unroll
        for (int r = 0; r < 8; ++r) {
            int mrow = m0 + r + 8 * hi;
            if (mrow < nrows) H[(size_t)mrow * COUT + t * 16 + lo] = acc[t][r];
        }
    }
}

// ---------------------------------------------------------------- edge scatter
// agg[dst[e], :] += H[src[e], :].  One wave per edge, C/32 floats per lane.
// H (51MB) and agg (51MB) both fit in the 192MB L2 -> L2-rate gather + atomics.
template <int C>
__global__ __launch_bounds__(256) void scatter_edges(const float* __restrict__ H,
                                                     const int* __restrict__ src,
                                                     const int* __restrict__ dst,
                                                     float* __restrict__ agg,
                                                     int nedges) {
    const int lane = threadIdx.x & 31;
    const int e = blockIdx.x * 8 + (threadIdx.x >> 5);
    if (e >= nedges) return;
    const int s = src[e];
    const int d = dst[e];
    constexpr int PER = C / 32;
    const float* hp = H + (size_t)s * C + lane * PER;
    float* ap = agg + (size_t)d * C + lane * PER;
#pragma unroll
    for (int j = 0; j < PER; ++j) unsafeAtomicAdd(ap + j, hp[j]);
}

// ---------------------------------------------------------------- finish layer
// out = relu?(agg * isi[node] + b[c])
template <int C, bool RELU>
__global__ __launch_bounds__(256) void finish_kernel(const float* __restrict__ agg,
                                                     const float* __restrict__ isi,
                                                     const float* __restrict__ b,
                                                     float* __restrict__ out,
                                                     int nnodes) {
    int i = blockIdx.x * 256 + threadIdx.x;
    if (i < nnodes * C) {
        int node = i / C;
        int c = i - node * C;
        float v = agg[i] * isi[node] + b[c];
        if (RELU) v = v > 0.0f ? v : 0.0f;
        out[i] = v;
    }
}

// ---------------------------------------------------------------- final: norm+bias+log_softmax
// One wave per node, 64 channels -> 2 per lane; wave-wide shfl reductions (wave32).
__global__ __launch_bounds__(256) void logsoftmax64_kernel(const float* __restrict__ agg,
                                                           const float* __restrict__ isi,
                                                           const float* __restrict__ b,
                                                           float* __restrict__ out,
                                                           int nnodes) {
    const int lane = threadIdx.x & 31;
    const int node = blockIdx.x * 8 + (threadIdx.x >> 5);
    if (node >= nnodes) return;
    const float si = isi[node];
    const int c0 = lane * 2;
    const size_t base = (size_t)node * 64 + c0;
    float v0 = agg[base + 0] * si + b[c0 + 0];
    float v1 = agg[base + 1] * si + b[c0 + 1];

    float m = fmaxf(v0, v1);
#pragma unroll
    for (int off = 16; off > 0; off >>= 1) m = fmaxf(m, __shfl_xor(m, off, 32));
    float sum = expf(v0 - m) + expf(v1 - m);
#pragma unroll
    for (int off = 16; off > 0; off >>= 1) sum += __shfl_xor(sum, off, 32);
    const float lse = m + logf(sum);

    out[base + 0] = v0 - lse;
    out[base + 1] = v1 - lse;
}

// ---------------------------------------------------------------- launcher
extern "C" void kernel_launch(void* const* d_in, const int* in_sizes, int n_in,
                              void* d_out, int out_size, void* d_ws, size_t ws_size,
                              hipStream_t stream) {
    const float* features = (const float*)d_in[0];  // [N, 128]
    const int* src = (const int*)d_in[1];           // [E]
    const int* dst = (const int*)d_in[2];           // [E]
    const float* W1 = (const float*)d_in[3];        // [128,128]
    const float* b1 = (const float*)d_in[4];        // [128]
    const float* W2 = (const float*)d_in[5];        // [128,128]
    const float* b2 = (const float*)d_in[6];        // [128]
    const float* W3 = (const float*)d_in[7];        // [128,64]
    const float* b3 = (const float*)d_in[8];        // [64]
    float* out = (float*)d_out;                     // [N, 64]

    float* ws = (float*)d_ws;
    float* iso = ws;                        // [N]   deg_out^-1/2
    float* isi = ws + N_NODES;              // [N]   deg_in^-1/2
    float* bufA = ws + 2 * N_NODES;         // [N,128]  h  (xW)
    float* bufB = bufA + (size_t)N_NODES * 128;  // [N,128]  agg / x_next

    const int NC128 = N_NODES * 128;
    const int NC64 = N_NODES * 64;
    const dim3 blk(256);
    const dim3 g_fill2N((2 * N_NODES + 255) / 256);
    const dim3 g_fill128((NC128 + 255) / 256);
    const dim3 g_fill64((NC64 + 255) / 256);
    const dim3 g_edges((N_EDGES + 255) / 256);
    const dim3 g_gemm((N_NODES + 127) / 128);
    const dim3 g_scat((N_EDGES + 7) / 8);
    const dim3 g_node8((N_NODES + 7) / 8);

    // --- degrees -> normalization scales ---
    fill_zero_kernel<<<g_fill2N, blk, 0, stream>>>(iso, 2 * N_NODES);
    degree_kernel<<<g_edges, blk, 0, stream>>>(src, dst, iso, isi, N_EDGES);
    invsqrt_kernel<<<g_fill2N, blk, 0, stream>>>(iso, 2 * N_NODES);

    // --- layer 1: h1 = (x0 * iso) @ W1 ; agg ; x1 = relu(agg*isi + b1) ---
    gemm_scaled_wmma<128, 128><<<g_gemm, blk, 0, stream>>>(features, iso, W1, bufA, N_NODES);
    fill_zero_kernel<<<g_fill128, blk, 0, stream>>>(bufB, NC128);
    scatter_edges<128><<<g_scat, blk, 0, stream>>>(bufA, src, dst, bufB, N_EDGES);
    finish_kernel<128, true><<<g_fill128, blk, 0, stream>>>(bufB, isi, b1, bufB, N_NODES);

    // --- layer 2 ---
    gemm_scaled_wmma<128, 128><<<g_gemm, blk, 0, stream>>>(bufB, iso, W2, bufA, N_NODES);
    fill_zero_kernel<<<g_fill128, blk, 0, stream>>>(bufB, NC128);
    scatter_edges<128><<<g_scat, blk, 0, stream>>>(bufA, src, dst, bufB, N_EDGES);
    finish_kernel<128, true><<<g_fill128, blk, 0, stream>>>(bufB, isi, b2, bufB, N_NODES);

    // --- layer 3 (128 -> 64) + log_softmax ---
    gemm_scaled_wmma<128, 64><<<g_gemm, blk, 0, stream>>>(bufB, iso, W3, bufA, N_NODES);
    fill_zero_kernel<<<g_fill64, blk, 0, stream>>>(bufB, NC64);
    scatter_edges<64><<<g_scat, blk, 0, stream>>>(bufA, src, dst, bufB, N_EDGES);
    logsoftmax64_kernel<<<g_node8, blk, 0, stream>>>(bufB, isi, b3, out, N_NODES);
}